// PretrainedGNNWrapper_86053964742818
// MI455X (gfx1250) — compile-verified
//
#include <hip/hip_runtime.h>
#include <hip/hip_bf16.h>

#define NUM_REL 7
#define NN 30000
#define EE 480000
#define D_IN 21
#define D_H 512
#define BN_EPS 1e-5f

typedef float __attribute__((ext_vector_type(2))) v2f;
typedef float __attribute__((ext_vector_type(8))) v8f;

// ---------------------------------------------------------------- utilities
__global__ void zero_kernel(float* __restrict__ p, size_t n) {
    size_t i = (size_t)blockIdx.x * blockDim.x + threadIdx.x;
    size_t stride = (size_t)gridDim.x * blockDim.x;
    for (; i < n; i += stride) p[i] = 0.0f;
}

// ------------------------------------------------- scatter-add (segment_sum)
// A[dst, rel*d + f] += x[src, f]   (A row stride = lda = 8*d)
__global__ void scatter_kernel(const float* __restrict__ x, int xstride,
                               const int* __restrict__ src,
                               const int* __restrict__ dst,
                               const int* __restrict__ rel,
                               float* __restrict__ A, int lda, int d) {
    size_t total = (size_t)EE * (size_t)d;
    size_t i = (size_t)blockIdx.x * blockDim.x + threadIdx.x;
    size_t stride = (size_t)gridDim.x * blockDim.x;
    for (; i < total; i += stride) {
        int e, f;
        if (d == 512) { e = (int)(i >> 9); f = (int)(i & 511); }
        else          { e = (int)(i / (size_t)d); f = (int)(i - (size_t)e * d); }
        int s = src[e], dt = dst[e], r = rel[e];
        float v = x[(size_t)s * xstride + f];
        atomicAdd(&A[(size_t)dt * lda + (size_t)r * d + f], v);
    }
}

// ------------------------------------- copy self-loop features into A[:,7d:8d]
__global__ void copyx_kernel(const float* __restrict__ x, int xstride,
                             float* __restrict__ A, int lda, int d) {
    size_t total = (size_t)NN * (size_t)d;
    size_t i = (size_t)blockIdx.x * blockDim.x + threadIdx.x;
    size_t stride = (size_t)gridDim.x * blockDim.x;
    for (; i < total; i += stride) {
        int row, f;
        if (d == 512) { row = (int)(i >> 9); f = (int)(i & 511); }
        else          { row = (int)(i / (size_t)d); f = (int)(i - (size_t)row * d); }
        A[(size_t)row * lda + 7 * d + f] = x[(size_t)row * xstride + f];
    }
}

// ------------------------------------------------ build Wc = [lin_w | self_w]
__global__ void buildw_kernel(const float* __restrict__ lin_w,
                              const float* __restrict__ self_w,
                              float* __restrict__ Wc, int d) {
    int K = 8 * d;
    size_t total = (size_t)D_H * (size_t)K;
    size_t i = (size_t)blockIdx.x * blockDim.x + threadIdx.x;
    size_t stride = (size_t)gridDim.x * blockDim.x;
    for (; i < total; i += stride) {
        int n = (int)(i / (size_t)K);
        int j = (int)(i - (size_t)n * K);
        float v = (j < 7 * d) ? lin_w[(size_t)n * (7 * d) + j]
                              : self_w[(size_t)n * d + (j - 7 * d)];
        Wc[i] = v;
    }
}

// ----------------------------------------------------------- WMMA fp32 GEMM
// out(N,512) = A(N,K) * Wc(512,K)^T + lin_b + self_b
// grid.x = N/16 row tiles; 256 threads = 8 waves; wave w -> cols [64w, 64w+64)
__global__ __launch_bounds__(256)
void gemm_wmma_kernel(const float* __restrict__ A, int lda,
                      const float* __restrict__ Wc, int K,
                      const float* __restrict__ lin_b,
                      const float* __restrict__ self_b,
                      float* __restrict__ out) {
    const int wave = threadIdx.x >> 5;
    const int lane = threadIdx.x & 31;
    const int lm = lane & 15;
    const int lh = lane >> 4;
    const int rowBase = blockIdx.x * 16;
    const int colBase = wave * 64;

    v8f c0 = {}, c1 = {}, c2 = {}, c3 = {};

    const float* arow = A + (size_t)(rowBase + lm) * lda + lh * 2;
    const float* w0 = Wc + (size_t)(colBase + 0 + lm) * K + lh * 2;
    const float* w1 = Wc + (size_t)(colBase + 16 + lm) * K + lh * 2;
    const float* w2 = Wc + (size_t)(colBase + 32 + lm) * K + lh * 2;
    const float* w3 = Wc + (size_t)(colBase + 48 + lm) * K + lh * 2;

    for (int kk = 0; kk < K; kk += 4) {
        // A fragment: lane holds A[m=lm][k = kk + lh*2 + {0,1}]
        v2f a = *reinterpret_cast<const v2f*>(arow + kk);
        // B fragments: lane holds Wc[n=col+lm][k = kk + lh*2 + {0,1}] == B^T
        v2f b0 = *reinterpret_cast<const v2f*>(w0 + kk);
        v2f b1 = *reinterpret_cast<const v2f*>(w1 + kk);
        v2f b2 = *reinterpret_cast<const v2f*>(w2 + kk);
        v2f b3 = *reinterpret_cast<const v2f*>(w3 + kk);
        c0 = __builtin_amdgcn_wmma_f32_16x16x4_f32(false, a, false, b0, (short)0, c0, false, false);
        c1 = __builtin_amdgcn_wmma_f32_16x16x4_f32(false, a, false, b1, (short)0, c1, false, false);
        c2 = __builtin_amdgcn_wmma_f32_16x16x4_f32(false, a, false, b2, (short)0, c2, false, false);
        c3 = __builtin_amdgcn_wmma_f32_16x16x4_f32(false, a, false, b3, (short)0, c3, false, false);
    }

    // Epilogue: C/D VGPR r -> row (r + lh*8), col lm within each 16-col tile
    #pragma unroll
    for (int t = 0; t < 4; ++t) {
        v8f c = (t == 0) ? c0 : (t == 1) ? c1 : (t == 2) ? c2 : c3;
        int col = colBase + t * 16 + lm;
        float bias = lin_b[col] + self_b[col];
        #pragma unroll
        for (int r = 0; r < 8; ++r) {
            int row = rowBase + r + lh * 8;
            out[(size_t)row * D_H + col] = c[r] + bias;
        }
    }
}

// ---------------------------------------------------- BN column statistics
// X is (N, 512) with row stride xstride; 512 threads/block, one col/thread
__global__ __launch_bounds__(512)
void stats_kernel(const float* __restrict__ X, int xstride,
                  float* __restrict__ sum, float* __restrict__ sumsq) {
    int c = threadIdx.x;
    float ls = 0.0f, lss = 0.0f;
    for (int r = blockIdx.x; r < NN; r += gridDim.x) {
        float v = X[(size_t)r * xstride + c];
        ls += v;
        lss += v * v;
    }
    atomicAdd(&sum[c], ls);
    atomicAdd(&sumsq[c], lss);
}

__global__ __launch_bounds__(512)
void finalize_stats_kernel(const float* __restrict__ sum,
                           const float* __restrict__ sumsq,
                           const float* __restrict__ g,
                           const float* __restrict__ b,
                           float* __restrict__ scale,
                           float* __restrict__ shift) {
    int c = threadIdx.x;
    float mu = sum[c] * (1.0f / NN);
    float var = sumsq[c] * (1.0f / NN) - mu * mu;
    float s = g[c] * rsqrtf(var + BN_EPS);
    scale[c] = s;
    shift[c] = b[c] - mu * s;
}

__global__ void apply_bn_kernel(const float* __restrict__ X, int xstride,
                                const float* __restrict__ scale,
                                const float* __restrict__ shift,
                                float* __restrict__ Y, int ystride, int relu) {
    size_t total = (size_t)NN * (size_t)D_H;
    size_t i = (size_t)blockIdx.x * blockDim.x + threadIdx.x;
    size_t stride = (size_t)gridDim.x * blockDim.x;
    for (; i < total; i += stride) {
        int row = (int)(i >> 9);
        int c = (int)(i & 511);
        float y = X[(size_t)row * xstride + c] * scale[c] + shift[c];
        if (relu) y = fmaxf(y, 0.0f);
        Y[(size_t)row * ystride + c] = y;
    }
}

// ---------------------------------------------------------------- launcher
extern "C" void kernel_launch(void* const* d_in, const int* in_sizes, int n_in,
                              void* d_out, int out_size, void* d_ws, size_t ws_size,
                              hipStream_t stream) {
    const float* x0 = (const float*)d_in[0];
    const int* src = (const int*)d_in[1];
    const int* dst = (const int*)d_in[2];
    const int* rel = (const int*)d_in[3];

    float* ws = (float*)d_ws;
    float* A     = ws;                         // N * 4096   (max layout)
    float* Wc    = A + (size_t)NN * 4096;      // 512 * 4096
    float* tmp   = Wc + (size_t)D_H * 4096;    // N * 512
    float* hrelu = tmp + (size_t)NN * D_H;     // N * 512
    float* sum   = hrelu + (size_t)NN * D_H;   // 512
    float* sumsq = sum + D_H;                  // 512
    float* scale = sumsq + D_H;                // 512
    float* shift = scale + D_H;                // 512

    float* out = (float*)d_out;                // (N, 1536)

    const float* xin = x0;
    int xstride = D_IN;

    for (int layer = 0; layer < 3; ++layer) {
        int d = (layer == 0) ? D_IN : D_H;
        int K = 8 * d;
        int base = 4 + layer * 8;
        const float* lin_w  = (const float*)d_in[base + 0];
        const float* self_w = (const float*)d_in[base + 1];
        const float* lin_b  = (const float*)d_in[base + 2];
        const float* self_b = (const float*)d_in[base + 3];
        const float* bn1_g  = (const float*)d_in[base + 4];
        const float* bn1_b  = (const float*)d_in[base + 5];
        const float* bn2_g  = (const float*)d_in[base + 6];
        const float* bn2_b  = (const float*)d_in[base + 7];

        // 1. zero A (upd region must start at 0 for the segment-sum)
        zero_kernel<<<8192, 256, 0, stream>>>(A, (size_t)NN * K);
        // 2. scatter-add messages into A[:, 0:7d]
        scatter_kernel<<<8192, 256, 0, stream>>>(xin, xstride, src, dst, rel, A, K, d);
        // 3. self features into A[:, 7d:8d]
        copyx_kernel<<<4096, 256, 0, stream>>>(xin, xstride, A, K, d);
        // 4. concatenated weights
        buildw_kernel<<<2048, 256, 0, stream>>>(lin_w, self_w, Wc, d);
        // 5. WMMA GEMM -> tmp
        gemm_wmma_kernel<<<NN / 16, 256, 0, stream>>>(A, K, Wc, K, lin_b, self_b, tmp);
        // 6. BN1 (+ReLU) -> hrelu
        zero_kernel<<<1, 1024, 0, stream>>>(sum, 2 * D_H);
        stats_kernel<<<240, 512, 0, stream>>>(tmp, D_H, sum, sumsq);
        finalize_stats_kernel<<<1, 512, 0, stream>>>(sum, sumsq, bn1_g, bn1_b, scale, shift);
        apply_bn_kernel<<<8192, 256, 0, stream>>>(tmp, D_H, scale, shift, hrelu, D_H, 1);
        // 7. BN2 -> d_out slice (also the next layer's input)
        zero_kernel<<<1, 1024, 0, stream>>>(sum, 2 * D_H);
        stats_kernel<<<240, 512, 0, stream>>>(hrelu, D_H, sum, sumsq);
        finalize_stats_kernel<<<1, 512, 0, stream>>>(sum, sumsq, bn2_g, bn2_b, scale, shift);
        apply_bn_kernel<<<8192, 256, 0, stream>>>(hrelu, D_H, scale, shift,
                                                  out + (size_t)layer * D_H, 3 * D_H, 0);

        xin = out + (size_t)layer * D_H;
        xstride = 3 * D_H;
    }
}